// SoftCrossEntropyLoss_1580547971294
// MI455X (gfx1250) — compile-verified
//
#include <hip/hip_runtime.h>
#include <hip/hip_bf16.h>

typedef __attribute__((ext_vector_type(4))) float v4f;
typedef __attribute__((ext_vector_type(2))) float v2f;
typedef __attribute__((ext_vector_type(8))) float v8f;

#define NROWS 8192
#define KDIM  32000
#define K4    (KDIM / 4)      // 8000 float4 per row, 16B-aligned (row pitch 128000 B)
#define BLK   256             // 8 waves (wave32)
#define NWAVES (BLK / 32)

// 32-lane sum reduction via V_WMMA_F32_16X16X4_F32 against an all-ones B.
// A layout (32-bit 16x4): lanes 0-15 VGPR0=K0, VGPR1=K1; lanes 16-31 VGPR0=K2, VGPR1=K3.
// With a = {v, 0}: D[M, n] = v_M + v_{M+16} for every column n (B==1 everywhere,
// so the result is independent of B's exact lane striping).
// C/D layout: lane n (0-15) holds M=0..7 in c[0..7]; lane n+16 holds M=8..15.
// Per-lane sum of 8 C regs + shfl_xor(16) => full 32-lane sum in every lane.
__device__ __forceinline__ float wave_sum_wmma(float v) {
    v2f a; a.x = v;    a.y = 0.0f;
    v2f b; b.x = 1.0f; b.y = 1.0f;
    v8f c = {};
    c = __builtin_amdgcn_wmma_f32_16x16x4_f32(false, a, false, b,
                                              (short)0, c, false, false);
    float s = c[0] + c[1] + c[2] + c[3] + c[4] + c[5] + c[6] + c[7];
    s += __shfl_xor(s, 16, 32);
    return s;
}

__global__ void __launch_bounds__(BLK)
sce_row_kernel(const float* __restrict__ x, const float* __restrict__ t,
               float* __restrict__ row_loss) {
    const int row = blockIdx.x;
    const size_t base = (size_t)row * (size_t)KDIM;
    const v4f* __restrict__ xp = (const v4f*)(x + base);
    const v4f* __restrict__ tp = (const v4f*)(t + base);
    const int tid = threadIdx.x;

    // Per-thread online-softmax state + dot/tsum accumulators.
    float m    = -3.402823466e+38f;
    float s    = 0.0f;
    float dot  = 0.0f;
    float tsum = 0.0f;

    #pragma unroll 2
    for (int i = tid; i < K4; i += BLK) {
        // Stream-once data: non-temporal b128 loads, don't pollute L2.
        v4f xv = __builtin_nontemporal_load(xp + i);
        v4f tv = __builtin_nontemporal_load(tp + i);

        float cm = fmaxf(fmaxf(xv.x, xv.y), fmaxf(xv.z, xv.w));
        if (cm > m) {                 // rare after warm-up: saves 1 exp/chunk
            s *= __expf(m - cm);
            m = cm;
        }
        s += __expf(xv.x - m);
        s += __expf(xv.y - m);
        s += __expf(xv.z - m);
        s += __expf(xv.w - m);

        dot  = fmaf(tv.x, xv.x, dot);
        dot  = fmaf(tv.y, xv.y, dot);
        dot  = fmaf(tv.z, xv.z, dot);
        dot  = fmaf(tv.w, xv.w, dot);
        tsum += (tv.x + tv.y) + (tv.z + tv.w);
    }

    // Wave-level combine of (m, s): online-softmax merge via wave32 shuffles.
    for (int off = 16; off > 0; off >>= 1) {
        float mo = __shfl_xor(m, off, 32);
        float so = __shfl_xor(s, off, 32);
        float nm = fmaxf(m, mo);
        s = s * __expf(m - nm) + so * __expf(mo - nm);
        m = nm;
    }

    // Wave-level sums of dot/tsum via WMMA (converged here => EXEC all 1s).
    dot  = wave_sum_wmma(dot);
    tsum = wave_sum_wmma(tsum);

    __shared__ float sm[NWAVES], ss[NWAVES], sd[NWAVES], st[NWAVES];
    const int wave = tid >> 5;
    const int lane = tid & 31;
    if (lane == 0) { sm[wave] = m; ss[wave] = s; sd[wave] = dot; st[wave] = tsum; }
    __syncthreads();

    if (tid == 0) {
        float M = sm[0], S = ss[0], D = sd[0], T = st[0];
        #pragma unroll
        for (int w = 1; w < NWAVES; ++w) {
            float mo = sm[w];
            float nm = fmaxf(M, mo);
            S = S * __expf(M - nm) + ss[w] * __expf(mo - nm);
            M = nm;
            D += sd[w];
            T += st[w];
        }
        const float logZ = M + logf(S);           // accurate log, once per row
        row_loss[row] = logZ * T - D;             // Σ t*(logZ - x)
    }
}

__global__ void __launch_bounds__(BLK)
mean_kernel(const float* __restrict__ row_loss, float* __restrict__ out) {
    __shared__ float red[NWAVES];
    float acc = 0.0f;
    for (int i = threadIdx.x; i < NROWS; i += BLK)
        acc += row_loss[i];
    for (int off = 16; off > 0; off >>= 1)
        acc += __shfl_xor(acc, off, 32);
    const int wave = threadIdx.x >> 5;
    const int lane = threadIdx.x & 31;
    if (lane == 0) red[wave] = acc;
    __syncthreads();
    if (threadIdx.x == 0) {
        float tot = 0.0f;
        #pragma unroll
        for (int w = 0; w < NWAVES; ++w) tot += red[w];
        out[0] = tot * (1.0f / (float)NROWS);
    }
}

extern "C" void kernel_launch(void* const* d_in, const int* in_sizes, int n_in,
                              void* d_out, int out_size, void* d_ws, size_t ws_size,
                              hipStream_t stream) {
    const float* x = (const float*)d_in[0];   // logits  [8192, 32000] f32
    const float* t = (const float*)d_in[1];   // targets [8192, 32000] f32
    float* row_loss = (float*)d_ws;           // 8192 f32 = 32 KB scratch

    sce_row_kernel<<<NROWS, BLK, 0, stream>>>(x, t, row_loss);
    mean_kernel<<<1, BLK, 0, stream>>>(row_loss, (float*)d_out);
}